// DifferenceComparisonLayer_62380105007208
// MI455X (gfx1250) — compile-verified
//
#include <hip/hip_runtime.h>
#include <stdint.h>

// DifferenceComparisonLayer for MI455X (gfx1250).
// Memory-bound streaming kernel: 235 MB total traffic -> ~10 us at 23.3 TB/s.
// Bulk path: 8 rows per block, 224 lanes x b128 per row, NT hints (footprint
// 235 MB > 192 MB L2). All 8 row-chunks are loaded back-to-back for 8
// outstanding b128 loads per wave, then stored back-to-back. a|b segments are
// DMA'd to LDS via the CDNA5 Tensor Data Mover (TENSORcnt/s_wait_tensorcnt);
// 3 lanes per row fuse the silu/cascade math and patch their output chunks.

#define DIM      896
#define CHUNKS   224          // DIM / 4
#define RPB      8            // rows per block
#define A_S      528          // a = x[528:536], b = x[536:544]
#define DCHUNK0  140          // floats 560..563  -> diff[0..3]
#define DCHUNK1  141          // floats 564..567  -> diff[4..7]
#define FCHUNK   142          // floats 568..570  -> eq, lt, gt (571 preserved)

typedef float    f32x4 __attribute__((ext_vector_type(4)));
typedef unsigned u32x4 __attribute__((ext_vector_type(4)));
typedef int      i32x4 __attribute__((ext_vector_type(4)));
typedef int      i32x8 __attribute__((ext_vector_type(8)));

#if defined(__has_builtin)
#if __has_builtin(__builtin_amdgcn_tensor_load_to_lds)
#define HAVE_TDM 1
#endif
#endif

__device__ __forceinline__ float silu_f(float v) {
    return v / (1.0f + __expf(-v));
}

// Patch one row-chunk v (chunk index c in [DCHUNK0, FCHUNK]) using the 16
// a|b floats staged in LDS for that row.
__device__ __forceinline__ void fixup_chunk(f32x4& v, const float* __restrict__ ab,
                                            int c) {
    float d[8], eqn[8], lt[8], gt[8];
#pragma unroll
    for (int i = 0; i < 8; ++i) {
        const float di = ab[i] - ab[8 + i];
        const float sd = 20.0f * di;
        d[i] = di;
        const float eq = silu_f(sd + 0.625f) * (0.625f - sd);
        eqn[i] = fminf(fmaxf(eq * (1.0f / 0.24f), 0.0f), 1.0f);
        lt[i]  = fmaxf(20.0f * silu_f(-sd - 0.625f), 0.0f);
        gt[i]  = fmaxf(20.0f * silu_f( sd - 0.625f), 0.0f);
    }
    if (c == DCHUNK0) {
        v[0] = d[0]; v[1] = d[1]; v[2] = d[2]; v[3] = d[3];
    } else if (c == DCHUNK1) {
        v[0] = d[4]; v[1] = d[5]; v[2] = d[6]; v[3] = d[7];
    } else {
        // casc[i] = prod_{j>i} eqn[j]; eq_final = prod of all eqn
        float casc = 1.0f, ltf = 0.0f, gtf = 0.0f;
#pragma unroll
        for (int i = 7; i >= 0; --i) {
            ltf  += casc * lt[i];
            gtf  += casc * gt[i];
            casc *= eqn[i];
        }
        v[0] = casc;
        v[1] = fminf(fmaxf(ltf * (1.0f / 20.0f), 0.0f), 1.0f);
        v[2] = fminf(fmaxf(gtf * (1.0f / 20.0f), 0.0f), 1.0f);
        // v[3] (x[..,571]) preserved from the load
    }
}

__global__ __launch_bounds__(CHUNKS)
void diffcmp_kernel(const float* __restrict__ x, float* __restrict__ out, int nrows) {
    __shared__ float shm[RPB * 16];   // per row: a[0..7], b[0..7]

    const int row0 = blockIdx.x * RPB;
    int rows_here = nrows - row0;
    if (rows_here > RPB) rows_here = RPB;
    if (rows_here <= 0) return;

#ifdef HAVE_TDM
    // One wave issues a single 2-D TDM transfer: tile 16 x rows_here, f32,
    // row stride = DIM elements, destination = shm (contiguous, no padding).
    if (threadIdx.x == 0) {
        const unsigned long long gaddr =
            (unsigned long long)(const char*)(x + (size_t)row0 * DIM + A_S);
        const unsigned lds_off = (unsigned)(uintptr_t)(void*)&shm[0];

        u32x4 g0;
        g0[0] = 1u;                                        // count=1, user descriptor
        g0[1] = lds_off;                                   // lds_addr (bytes)
        g0[2] = (unsigned)(gaddr & 0xFFFFFFFFull);         // global_addr[31:0]
        g0[3] = (unsigned)((gaddr >> 32) & 0x1FFFFFFull)   // global_addr[56:32]
              | (2u << 30);                                // type = 2 ("image")

        i32x8 g1;
        g1[0] = (2 << 16);                                 // data_size = 4 bytes
        g1[1] = (int)(16u << 16);                          // tensor_dim0 = 16
        g1[2] = (int)((unsigned)rows_here << 16);          // tensor_dim1 = rows_here
        g1[3] = (int)(16u << 16);                          // tile_dim0 = 16
        g1[4] = rows_here;                                 // tile_dim1; tile_dim2 = 0
        g1[5] = DIM;                                       // tensor_dim0_stride = 896
        g1[6] = 0;
        g1[7] = 0;

        i32x4 z4 = (i32x4)0;
#if defined(__clang_major__) && __clang_major__ >= 23
        i32x8 z8 = (i32x8)0;
        __builtin_amdgcn_tensor_load_to_lds(g0, g1, z4, z4, z8, 0);
#else
        __builtin_amdgcn_tensor_load_to_lds(g0, g1, z4, z4, 0);
#endif
        __builtin_amdgcn_s_wait_tensorcnt(0);
    }
#else
    // Fallback: 4 lanes per row stage the 16-float a|b segment into LDS.
    if ((int)threadIdx.x < rows_here * 4) {
        const int r = threadIdx.x >> 2, q = threadIdx.x & 3;
        const f32x4* s = (const f32x4*)(x + (size_t)(row0 + r) * DIM + A_S);
        ((f32x4*)shm)[threadIdx.x] = s[q];
    }
#endif
    __syncthreads();

    const int c = threadIdx.x;    // b128 chunk index within the row
    const bool special = (c >= DCHUNK0) && (c <= FCHUNK);

    if (rows_here == RPB) {
        // Hot path (always taken for the 32768-row problem): batch all loads,
        // patch, batch all stores -> 8 outstanding b128 loads per wave.
        f32x4 v[RPB];
#pragma unroll
        for (int r = 0; r < RPB; ++r) {
            const f32x4* src = (const f32x4*)(x + (size_t)(row0 + r) * DIM);
            v[r] = __builtin_nontemporal_load(&src[c]);
        }
        if (special) {
#pragma unroll 1   // keep the ~heavy transcendental body out of I-cache x8
            for (int r = 0; r < RPB; ++r)
                fixup_chunk(v[r], &shm[r * 16], c);
        }
#pragma unroll
        for (int r = 0; r < RPB; ++r) {
            f32x4* dst = (f32x4*)(out + (size_t)(row0 + r) * DIM);
            __builtin_nontemporal_store(v[r], &dst[c]);
        }
    } else {
        // Generic tail (unused at this problem size).
        for (int r = 0; r < rows_here; ++r) {
            const f32x4* src = (const f32x4*)(x   + (size_t)(row0 + r) * DIM);
            f32x4*       dst = (f32x4*)      (out + (size_t)(row0 + r) * DIM);
            f32x4 v = __builtin_nontemporal_load(&src[c]);
            if (special) fixup_chunk(v, &shm[r * 16], c);
            __builtin_nontemporal_store(v, &dst[c]);
        }
    }
}

extern "C" void kernel_launch(void* const* d_in, const int* in_sizes, int n_in,
                              void* d_out, int out_size, void* d_ws, size_t ws_size,
                              hipStream_t stream) {
    (void)n_in; (void)out_size; (void)d_ws; (void)ws_size;
    const float* x = (const float*)d_in[0];   // (8, 4096, 896) f32; weights are baked
    float* out = (float*)d_out;
    const int nrows = in_sizes[0] / DIM;      // 32768
    const int nblocks = (nrows + RPB - 1) / RPB;
    diffcmp_kernel<<<nblocks, CHUNKS, 0, stream>>>(x, out, nrows);
}